// MixedRON_79989470920785
// MI455X (gfx1250) — compile-verified
//
#include <hip/hip_runtime.h>

#define N_INP    32
#define N_HID    2048
#define PORTION  1024
#define BATCH    128
#define T_STEPS  200
#define DT_      0.042f
#define THRESH   0.5f

#define KCHUNK   256
#define NCHUNKS  (N_HID / KCHUNK)      // 8
#define LDS_K    (KCHUNK + 4)          // pad 4 floats: bank = (4*col + k) % 64

typedef __attribute__((ext_vector_type(2))) float v2f;
typedef __attribute__((ext_vector_type(4))) float v4f;
typedef __attribute__((ext_vector_type(8))) float v8f;

// One recurrence step: f = tanh(x_t @ x2h + hy_{t-1} @ h2h), then elementwise
// harmonic (cols < 1024) / spiking (cols >= 1024) updates.
//
// Grid: 128 blocks x 256 threads (8 wave32/block).
//  - block  -> one 16-wide column tile of h2h / f   (n_base = blockIdx.x*16)
//  - wave w -> one 16-row batch tile                (m_base = w*16)
// The block stages its B column tile through LDS (transposed, double-buffered
// 256-row K chunks) so each h2h element is read from L2 exactly once per step
// and all 8 waves consume it via conflict-free ds_load_b64.
// State is read from the t-1 slice of the output tensors (zeros at t==0).
__global__ __launch_bounds__(256) void mixed_ron_step(
    const float* __restrict__ x,       // (BATCH, T, N_INP)
    const float* __restrict__ x2h,     // (N_INP, N_HID)
    const float* __restrict__ h2h,     // (N_HID, N_HID)
    const float* __restrict__ gamma_,  // (N_HID), first PORTION used
    const float* __restrict__ eps_,    // (N_HID), first PORTION used
    float* __restrict__ out,           // hy_all | hz_all | hyu_all | spk_all
    int t)
{
    __shared__ float ldsB[2][16][LDS_K];   // [buf][local col][k in chunk]

    const int tid    = threadIdx.x;
    const int wave   = tid >> 5;
    const int lane   = tid & 31;
    const int lane15 = lane & 15;
    const int khalf  = (lane >> 4) * 2;    // 0 (lanes 0-15) or 2 (lanes 16-31)

    const int m_base = wave * 16;          // 8 waves cover 128 batch rows
    const int n_base = blockIdx.x * 16;    // 128 blocks cover 2048 cols

    float* hy_all  = out;
    float* hz_all  = out     + (size_t)T_STEPS * BATCH * N_HID;
    float* hyu_all = hz_all  + (size_t)T_STEPS * BATCH * PORTION;
    float* spk_all = hyu_all + (size_t)T_STEPS * BATCH * PORTION;

    v8f acc = {};

    // ---- input GEMM: x_t (16x32) @ x2h (32x16), K = 32 (direct, tiny) ----
    {
        const float* xrow  = x + (size_t)(m_base + lane15) * (T_STEPS * N_INP)
                               + (size_t)t * N_INP;
        const float* xbcol = x2h + n_base + lane15;
        #pragma unroll
        for (int k = 0; k < N_INP; k += 4) {
            v2f a = *(const v2f*)(xrow + k + khalf);
            v2f b;
            b.x = xbcol[(size_t)(k + khalf + 0) * N_HID];
            b.y = xbcol[(size_t)(k + khalf + 1) * N_HID];
            acc = __builtin_amdgcn_wmma_f32_16x16x4_f32(
                      false, a, false, b, (short)0, acc, false, false);
        }
    }

    // ---- recurrent GEMM: hy_{t-1} (16x2048) @ h2h (2048x16), K = 2048 ----
    if (t > 0) {
        const float* arow = hy_all + (size_t)(t - 1) * BATCH * N_HID
                                   + (size_t)(m_base + lane15) * N_HID;

        // Cooperative B staging: thread -> (row srow+64p, quad of 4 cols).
        const int    srow = tid >> 2;                  // 0..63
        const int    qc   = (tid & 3) * 4;             // local col group 0/4/8/12
        const float* gB   = h2h + (size_t)n_base + (size_t)qc;

        v4f pre[4];
        #pragma unroll
        for (int p = 0; p < 4; ++p)                    // prefetch chunk 0
            pre[p] = *(const v4f*)(gB + (size_t)(srow + p * 64) * N_HID);
        #pragma unroll
        for (int p = 0; p < 4; ++p) {                  // transpose into LDS
            const int r = srow + p * 64;
            ldsB[0][qc + 0][r] = pre[p].x;
            ldsB[0][qc + 1][r] = pre[p].y;
            ldsB[0][qc + 2][r] = pre[p].z;
            ldsB[0][qc + 3][r] = pre[p].w;
        }
        __syncthreads();

        for (int c = 0; c < NCHUNKS; ++c) {
            if (c + 1 < NCHUNKS) {                     // prefetch next chunk
                const float* gBc = gB + (size_t)(c + 1) * KCHUNK * N_HID;
                #pragma unroll
                for (int p = 0; p < 4; ++p)
                    pre[p] = *(const v4f*)(gBc + (size_t)(srow + p * 64) * N_HID);
            }

            const float* aK = arow + c * KCHUNK;
            const int    cb = c & 1;
            #pragma unroll 8
            for (int k = 0; k < KCHUNK; k += 4) {
                v2f a = *(const v2f*)(aK + k + khalf);
                v2f b = *(const v2f*)&ldsB[cb][lane15][k + khalf];
                acc = __builtin_amdgcn_wmma_f32_16x16x4_f32(
                          false, a, false, b, (short)0, acc, false, false);
            }

            if (c + 1 < NCHUNKS) {                     // fill other buffer
                const int nb = (c + 1) & 1;
                #pragma unroll
                for (int p = 0; p < 4; ++p) {
                    const int r = srow + p * 64;
                    ldsB[nb][qc + 0][r] = pre[p].x;
                    ldsB[nb][qc + 1][r] = pre[p].y;
                    ldsB[nb][qc + 2][r] = pre[p].z;
                    ldsB[nb][qc + 3][r] = pre[p].w;
                }
            }
            __syncthreads();
        }
    }

    // ---- epilogue: per-lane 8 elements of the 16x16 C tile ----
    // C layout: VGPR r -> row (r + 8*(lane>=16)), col = lane&15.
    const int    col     = n_base + lane15;
    const int    row_off = (lane >> 4) * 8;
    const size_t o_hy    = (size_t)t * BATCH * N_HID;
    const size_t o_p     = (size_t)t * BATCH * PORTION;
    const size_t p_hy    = (size_t)(t - 1) * BATCH * N_HID;
    const size_t p_p     = (size_t)(t - 1) * BATCH * PORTION;

    if (col < PORTION) {
        const float g = gamma_[col];
        const float e = eps_[col];
        #pragma unroll
        for (int r = 0; r < 8; ++r) {
            const int   b   = m_base + row_off + r;
            const float f   = tanhf(acc[r]);
            const float hyp = (t > 0) ? hy_all[p_hy + (size_t)b * N_HID + col] : 0.0f;
            const float hzp = (t > 0) ? hz_all[p_p + (size_t)b * PORTION + col] : 0.0f;
            const float hzn = hzp + DT_ * (f - g * hyp - e * hzp);
            const float hyn = hyp + DT_ * hzn;
            hy_all[o_hy + (size_t)b * N_HID + col]   = hyn;
            hz_all[o_p  + (size_t)b * PORTION + col] = hzn;
        }
    } else {
        const int pc = col - PORTION;
        #pragma unroll
        for (int r = 0; r < 8; ++r) {
            const int   b  = m_base + row_off + r;
            const float f  = tanhf(acc[r]);
            float       u  = (t > 0) ? hy_all[p_hy + (size_t)b * N_HID + col] : 0.0f;
            const float sp = (u > THRESH) ? 1.0f : 0.0f;
            u = (sp == 1.0f) ? 0.0f : u;
            const float hyu = u + DT_ * (f - u);
            hy_all[o_hy + (size_t)b * N_HID + col]  = hyu;
            hyu_all[o_p + (size_t)b * PORTION + pc] = hyu;
            spk_all[o_p + (size_t)b * PORTION + pc] = sp;
        }
    }
}

extern "C" void kernel_launch(void* const* d_in, const int* in_sizes, int n_in,
                              void* d_out, int out_size, void* d_ws, size_t ws_size,
                              hipStream_t stream) {
    const float* x      = (const float*)d_in[0];
    const float* x2h    = (const float*)d_in[1];
    const float* h2h    = (const float*)d_in[2];
    const float* gamma_ = (const float*)d_in[3];
    const float* eps_   = (const float*)d_in[4];
    float*       out    = (float*)d_out;

    // 200 dependent recurrence steps, ordered on the stream.
    for (int t = 0; t < T_STEPS; ++t) {
        mixed_ron_step<<<128, 256, 0, stream>>>(x, x2h, h2h, gamma_, eps_, out, t);
    }
}